// PQMatcher_24352464569883
// MI455X (gfx1250) — compile-verified
//
#include <hip/hip_runtime.h>

typedef __attribute__((ext_vector_type(2))) float v2f;
typedef __attribute__((ext_vector_type(8))) float v8f;

#define LP  512
#define LQ  256
#define BB  32
#define IN  512
#define H   150
#define HP  160     // padded H
#define NH3 464     // padded 3H (450 -> 29*16)
#define KPV 672     // 512 + 160
#define NJ  1024    // used Wg output columns (rows 1024..2047 of Wg)

// ---- workspace layout (float offsets) ----
#define OFF_WPV   0                                   // [HP][KPV]  row-major
#define OFF_WQS   (OFF_WPV + HP*KPV)                  // [HP][IN]   row-major
#define OFF_WQUQT (OFF_WQS + HP*IN)                   // [BB][HP][LQ]
#define OFF_WGCAT (OFF_WQUQT + BB*HP*LQ)              // [NJ][NJ]   row-major
#define OFF_WIH   (OFF_WGCAT + NJ*NJ)                 // [NH3][NJ]  row-major
#define OFF_WHH   (OFF_WIH + NH3*NJ)                  // [NH3][HP]  row-major
#define OFF_BIH   (OFF_WHH + NH3*HP)                  // [NH3]
#define OFF_BHH   (OFF_BIH + NH3)                     // [NH3]
#define OFF_V     (OFF_BHH + NH3)                     // v state [BB][HP] (padded)
#define OFF_C     (OFF_V + BB*HP)                     // c half  [BB][IN]
#define OFF_CC    (OFF_C + BB*IN)                     // gated c_ [BB][NJ]
#define OFF_GH    (OFF_CC + BB*NJ)                    // [BB][NH3]
#define OFF_GI    (OFF_GH + BB*NH3)                   // [BB][NH3]
#define OFF_BAR   (OFF_GI + BB*NH3)                   // 2 x u32

__device__ __forceinline__ float sigf(float x) {
  return 1.0f / (1.0f + __expf(-x));
}
__device__ __forceinline__ float tanh_fast(float x) {
  float ax = fabsf(x);
  float e = __expf(-2.0f * ax);
  float t = (1.0f - e) / (1.0f + e);
  return copysignf(t, x);
}

__device__ __forceinline__ v8f wmma4(v2f a, v2f b, v8f c) {
  // D = A(16x4 f32) * B(4x16 f32) + C(16x16 f32)
  return __builtin_amdgcn_wmma_f32_16x16x4_f32(false, a, false, b, (short)0, c,
                                               false, false);
}

// ---------------- prep: fold/pad weights, init state ----------------
__global__ void pqm_prep_kernel(const float* __restrict__ Wp, const float* __restrict__ Wq,
                                const float* __restrict__ Wv, const float* __restrict__ Wg,
                                const float* __restrict__ W_ih, const float* __restrict__ W_hh,
                                const float* __restrict__ b_ih, const float* __restrict__ b_hh,
                                const float* __restrict__ v0, float* __restrict__ ws) {
  const long NT1 = HP * KPV;   // Wpv = [Wp_sum | Wv] padded
  const long NT2 = HP * IN;    // Wq_sum padded
  const long NT3 = (long)NJ * NJ;    // Wg_cat (rows 1024.. folded)
  const long NT4 = (long)NH3 * NJ;   // W_ih padded
  const long NT5 = (long)NH3 * HP;   // W_hh padded
  const long NT6 = NH3, NT7 = NH3, NT8 = BB * HP;
  const long total = NT1 + NT2 + NT3 + NT4 + NT5 + NT6 + NT7 + NT8;
  for (long idx = (long)blockIdx.x * blockDim.x + threadIdx.x; idx < total;
       idx += (long)gridDim.x * blockDim.x) {
    long t = idx;
    if (t < NT1) {
      int h = (int)(t / KPV), k = (int)(t % KPV);
      float v = 0.f;
      if (h < H) {
        if (k < IN) v = Wp[(size_t)h * 2 * IN + k] + Wp[(size_t)h * 2 * IN + IN + k];
        else { int j = k - IN; if (j < H) v = Wv[(size_t)h * H + j]; }
      }
      ws[OFF_WPV + t] = v; continue;
    }
    t -= NT1;
    if (t < NT2) {
      int h = (int)(t / IN), k = (int)(t % IN);
      ws[OFF_WQS + t] = (h < H) ? (Wq[(size_t)h * 2 * IN + k] + Wq[(size_t)h * 2 * IN + IN + k]) : 0.f;
      continue;
    }
    t -= NT2;
    if (t < NT3) {
      int j = (int)(t / NJ), k = (int)(t % NJ);
      const float* row = Wg + (size_t)(NJ + j) * (4 * IN);
      ws[OFF_WGCAT + t] = (k < IN) ? (row[k] + row[IN + k])
                                   : (row[2 * IN + (k - IN)] + row[3 * IN + (k - IN)]);
      continue;
    }
    t -= NT3;
    if (t < NT4) {
      int n = (int)(t / NJ), k = (int)(t % NJ);
      ws[OFF_WIH + t] = (n < 3 * H) ? W_ih[(size_t)n * NJ + k] : 0.f; continue;
    }
    t -= NT4;
    if (t < NT5) {
      int n = (int)(t / HP), j = (int)(t % HP);
      ws[OFF_WHH + t] = (n < 3 * H && j < H) ? W_hh[(size_t)n * H + j] : 0.f; continue;
    }
    t -= NT5;
    if (t < NT6) { ws[OFF_BIH + t] = (t < 3 * H) ? b_ih[t] : 0.f; continue; }
    t -= NT6;
    if (t < NT7) { ws[OFF_BHH + t] = (t < 3 * H) ? b_hh[t] : 0.f; continue; }
    t -= NT7;
    { int b = (int)(t / HP), j = (int)(t % HP);
      ws[OFF_V + t] = (j < H) ? v0[(size_t)b * H + j] : 0.f; }
  }
  if (blockIdx.x == 0 && threadIdx.x == 0) {
    unsigned* bar = (unsigned*)(ws + OFF_BAR);
    bar[0] = 0u; bar[1] = 0u;
  }
}

// -------- precompute WqUq^T via f32 WMMA: (8192x512)@(512x160)^T --------
__global__ void pqm_wquq_gemm_kernel(const float* __restrict__ uq, float* __restrict__ ws) {
  const float* WqS = ws + OFF_WQS;
  float* outT = ws + OFF_WQUQT;
  const int lane = threadIdx.x & 31;
  const int tile = blockIdx.x * (blockDim.x >> 5) + (threadIdx.x >> 5);
  const int MT = (LQ * BB) / 16;  // 512
  const int NT = HP / 16;         // 10
  if (tile >= MT * NT) return;
  const int mt = tile / NT, nt = tile % NT;
  const int r = lane & 15, hi = lane >> 4;
  const float* ap = uq + (size_t)(mt * 16 + r) * IN + 2 * hi;
  const float* bp = WqS + (size_t)(nt * 16 + r) * IN + 2 * hi;
  v8f acc = {};
  for (int k = 0; k < IN; k += 4) {
    v2f a = *(const v2f*)(ap + k);
    v2f b = *(const v2f*)(bp + k);
    acc = wmma4(a, b, acc);
  }
  const int h = nt * 16 + r;
#pragma unroll
  for (int j = 0; j < 8; ++j) {
    int M = mt * 16 + j + hi * 8;   // flattened q*32+b
    int b = M & (BB - 1), q = M >> 5;
    outT[((size_t)b * HP + h) * LQ + q] = acc[j];   // [b][h][q] for coalesced scan reads
  }
}

// ---------------- device-scope grid barrier ----------------
__device__ __forceinline__ void grid_barrier(volatile unsigned* cnt, volatile unsigned* gen,
                                             unsigned nb) {
  __threadfence();       // release my stores to device scope
  __syncthreads();
  if (threadIdx.x == 0) {
    unsigned g = *gen;
    unsigned arrived = atomicAdd((unsigned*)cnt, 1u);
    if (arrived == nb - 1u) {
      *cnt = 0u;
      __threadfence();
      atomicAdd((unsigned*)gen, 1u);
    } else {
      while (*gen == g) { __builtin_amdgcn_s_sleep(2); }
    }
  }
  __syncthreads();
  __threadfence();       // acquire others' stores
}

// ---------------- persistent recurrent scan ----------------
__global__ __launch_bounds__(1024, 1) void pqm_scan_kernel(
    const float* __restrict__ up, const float* __restrict__ uq,
    const float* __restrict__ V, float* __restrict__ ws, float* __restrict__ out) {
  __shared__ float sh_up[IN];
  __shared__ float sh_v[HP];
  __shared__ float sh_base[HP];
  __shared__ float sh_s[LQ];
  __shared__ float sh_red[1024];
  __shared__ float sh_V[H + 2];

  const int tid = threadIdx.x;
  const int lane = tid & 31;
  const int wv = tid >> 5;
  const int b = blockIdx.x;           // one batch row per block for phase 1
  const int gw = b * 32 + wv;         // global wave id (0..1023)

  float* v_ws = ws + OFF_V;
  float* c_ws = ws + OFF_C;
  float* cc_ws = ws + OFF_CC;
  float* gh_ws = ws + OFF_GH;
  float* gi_ws = ws + OFF_GI;
  const float* Wpv = ws + OFF_WPV;
  const float* wqT = ws + OFF_WQUQT + (size_t)b * HP * LQ;
  const float* WgCat = ws + OFF_WGCAT;
  const float* Wih = ws + OFF_WIH;
  const float* Whh = ws + OFF_WHH;
  const float* bih = ws + OFF_BIH;
  const float* bhh = ws + OFF_BHH;
  volatile unsigned* cnt = (volatile unsigned*)(ws + OFF_BAR);
  volatile unsigned* gen = cnt + 1;

  if (tid < H) sh_V[tid] = V[(size_t)b * H + tid];

  for (int p = 0; p < LP; ++p) {
    // ======== Phase 1 (per-b): base, s=tanh-reduce, softmax, context c ========
    const float* uprow = up + ((size_t)p * BB + b) * IN;
    if (tid < IN) sh_up[tid] = uprow[tid];
    else if (tid < IN + HP) sh_v[tid - IN] = v_ws[(size_t)b * HP + (tid - IN)];
    if (p + 1 < LP && tid < 32)
      __builtin_prefetch(up + ((size_t)(p + 1) * BB + b) * IN + tid * 16, 0, 0);
    __syncthreads();

    // base[h] = sum_k [up|v][k] * Wpv[h][k]   (wave-per-output, K=672)
    for (int rr = 0; rr < HP / 32; ++rr) {
      int h = rr * 32 + wv;
      const float* wr = Wpv + (size_t)h * KPV;
      float acc = 0.f;
      for (int k = lane; k < KPV; k += 32) {
        float x = (k < IN) ? sh_up[k] : sh_v[k - IN];
        acc += x * wr[k];
      }
      acc += __shfl_xor(acc, 16); acc += __shfl_xor(acc, 8);
      acc += __shfl_xor(acc, 4);  acc += __shfl_xor(acc, 2);
      acc += __shfl_xor(acc, 1);
      if (lane == 0) sh_base[h] = acc;
    }
    __syncthreads();

    // s[q] = sum_h tanh(WqUq[b][h][q] + base[h]) * V[b][h]   (4 threads per q)
    {
      int q = tid & (LQ - 1);
      int sub = tid >> 8;   // 0..3
      float acc = 0.f;
      for (int h = sub; h < H; h += 4)
        acc += tanh_fast(wqT[(size_t)h * LQ + q] + sh_base[h]) * sh_V[h];
      sh_red[sub * LQ + q] = acc;
    }
    __syncthreads();
    if (tid < LQ)
      sh_s[tid] = sh_red[tid] + sh_red[LQ + tid] + sh_red[2 * LQ + tid] + sh_red[3 * LQ + tid];
    __syncthreads();
    if (wv == 0) {   // softmax over q (wave 0)
      float m = -3.0e38f;
      for (int j = 0; j < LQ / 32; ++j) m = fmaxf(m, sh_s[lane + 32 * j]);
      m = fmaxf(m, __shfl_xor(m, 16)); m = fmaxf(m, __shfl_xor(m, 8));
      m = fmaxf(m, __shfl_xor(m, 4));  m = fmaxf(m, __shfl_xor(m, 2));
      m = fmaxf(m, __shfl_xor(m, 1));
      float s = 0.f;
      for (int j = 0; j < LQ / 32; ++j) {
        float e = __expf(sh_s[lane + 32 * j] - m);
        sh_s[lane + 32 * j] = e; s += e;
      }
      s += __shfl_xor(s, 16); s += __shfl_xor(s, 8); s += __shfl_xor(s, 4);
      s += __shfl_xor(s, 2);  s += __shfl_xor(s, 1);
      float inv = 1.0f / s;
      for (int j = 0; j < LQ / 32; ++j) sh_s[lane + 32 * j] *= inv;
    }
    __syncthreads();
    // c[i] = sum_q a[q] * uq[q][b][i]   (2 threads per i)
    {
      int i = tid & (IN - 1);
      int sub = tid >> 9;   // 0..1
      float acc = 0.f;
      const float* uqb = uq + (size_t)b * IN + (size_t)sub * 128 * BB * IN + i;
      for (int q = 0; q < LQ / 2; ++q)
        acc += sh_s[sub * 128 + q] * uqb[(size_t)q * BB * IN];
      sh_red[sub * IN + i] = acc;
    }
    __syncthreads();
    if (tid < IN) c_ws[(size_t)b * IN + tid] = sh_red[tid] + sh_red[IN + tid];
    grid_barrier(cnt, gen, BB);

    // ======== Phase 2 (WMMA): pre = [up|c] @ WgCat^T -> sigmoid*c ; gh = v @ Whh^T ========
    {
      const int r = lane & 15, hi = lane >> 4;
      if (gw < 128) {                       // E1: 2x64 tiles of (32 x 1024), K=1024
        int mt = gw >> 6, nt = gw & 63;
        int m = mt * 16 + r;
        const float* aup = up + ((size_t)p * BB + m) * IN + 2 * hi;
        const float* ac = c_ws + (size_t)m * IN + 2 * hi;
        const float* bp = WgCat + (size_t)(nt * 16 + r) * NJ + 2 * hi;
        v8f acc = {};
        for (int k = 0; k < IN; k += 4) {
          v2f a = *(const v2f*)(aup + k);
          v2f bm = *(const v2f*)(bp + k);
          acc = wmma4(a, bm, acc);
        }
        for (int k = 0; k < IN; k += 4) {
          v2f a = *(const v2f*)(ac + k);
          v2f bm = *(const v2f*)(bp + IN + k);
          acc = wmma4(a, bm, acc);
        }
        int col = nt * 16 + r;
#pragma unroll
        for (int j = 0; j < 8; ++j) {
          int row = mt * 16 + j + hi * 8;
          float g = sigf(acc[j]);
          cc_ws[(size_t)row * NJ + col] = g * c_ws[(size_t)row * IN + (col & (IN - 1))];
        }
      } else if (gw < 128 + 58) {           // gh: 2x29 tiles of (32 x 464), K=160
        int t2 = gw - 128;
        int mt = t2 / 29, nt = t2 % 29;
        const float* ap = v_ws + (size_t)(mt * 16 + r) * HP + 2 * hi;
        const float* bp = Whh + (size_t)(nt * 16 + r) * HP + 2 * hi;
        v8f acc = {};
        for (int k = 0; k < HP; k += 4) {
          v2f a = *(const v2f*)(ap + k);
          v2f bm = *(const v2f*)(bp + k);
          acc = wmma4(a, bm, acc);
        }
        int col = nt * 16 + r;
        float bias = bhh[col];
#pragma unroll
        for (int j = 0; j < 8; ++j) {
          int row = mt * 16 + j + hi * 8;
          gh_ws[(size_t)row * NH3 + col] = acc[j] + bias;
        }
      }
    }
    grid_barrier(cnt, gen, BB);

    // ======== Phase 3 (WMMA): gi = c_ @ Wih^T + bih  (2x29 tiles, K=1024) ========
    if (gw < 58) {
      const int r = lane & 15, hi = lane >> 4;
      int mt = gw / 29, nt = gw % 29;
      const float* ap = cc_ws + (size_t)(mt * 16 + r) * NJ + 2 * hi;
      const float* bp = Wih + (size_t)(nt * 16 + r) * NJ + 2 * hi;
      v8f acc = {};
      for (int k = 0; k < NJ; k += 4) {
        v2f a = *(const v2f*)(ap + k);
        v2f bm = *(const v2f*)(bp + k);
        acc = wmma4(a, bm, acc);
      }
      int col = nt * 16 + r;
      float bias = bih[col];
#pragma unroll
      for (int j = 0; j < 8; ++j) {
        int row = mt * 16 + j + hi * 8;
        gi_ws[(size_t)row * NH3 + col] = acc[j] + bias;
      }
    }
    grid_barrier(cnt, gen, BB);

    // ======== Phase 4: GRU elementwise update ========
    {
      int gt = b * 1024 + tid;
      if (gt < BB * H) {
        int b2 = gt / H, h = gt - b2 * H;
        const float* gi = gi_ws + (size_t)b2 * NH3;
        const float* gh = gh_ws + (size_t)b2 * NH3;
        float rg = sigf(gi[h] + gh[h]);
        float zg = sigf(gi[H + h] + gh[H + h]);
        float ng = tanh_fast(gi[2 * H + h] + rg * gh[2 * H + h]);
        float vn = (1.0f - zg) * ng + zg * v_ws[(size_t)b2 * HP + h];
        v_ws[(size_t)b2 * HP + h] = vn;
        out[(size_t)p * (BB * H) + gt] = vn;
      }
    }
    grid_barrier(cnt, gen, BB);
  }
}

extern "C" void kernel_launch(void* const* d_in, const int* in_sizes, int n_in,
                              void* d_out, int out_size, void* d_ws, size_t ws_size,
                              hipStream_t stream) {
  const float* up = (const float*)d_in[0];
  const float* uq = (const float*)d_in[1];
  const float* v0 = (const float*)d_in[2];
  const float* V = (const float*)d_in[3];
  const float* Wp = (const float*)d_in[4];
  const float* Wq = (const float*)d_in[5];
  const float* Wv = (const float*)d_in[6];
  const float* Wg = (const float*)d_in[7];
  const float* W_ih = (const float*)d_in[8];
  const float* W_hh = (const float*)d_in[9];
  const float* b_ih = (const float*)d_in[10];
  const float* b_hh = (const float*)d_in[11];
  float* out = (float*)d_out;
  float* ws = (float*)d_ws;
  (void)in_sizes; (void)n_in; (void)out_size; (void)ws_size;

  pqm_prep_kernel<<<512, 256, 0, stream>>>(Wp, Wq, Wv, Wg, W_ih, W_hh, b_ih, b_hh, v0, ws);
  pqm_wquq_gemm_kernel<<<640, 256, 0, stream>>>(uq, ws);
  pqm_scan_kernel<<<BB, 1024, 0, stream>>>(up, uq, V, ws, out);
}